// C51_69776038691350
// MI455X (gfx1250) — compile-verified
//
#include <hip/hip_runtime.h>

#define N_ATOMS      51
#define VMIN_C       (-10.0f)
#define VMAX_C       (10.0f)
#define GAMMA_C      (0.99f)
#define ACTION_DIM   4
#define WAVES_PER_BLOCK 8
#define BLOCK_THREADS   256
#define NBLK         2048   // partials count; multiple of 64 for the WMMA reduce

typedef __attribute__((ext_vector_type(2))) float v2f;
typedef __attribute__((ext_vector_type(8))) float v8f;

__device__ __forceinline__ float wave_max_f(float v) {
#pragma unroll
  for (int m = 16; m > 0; m >>= 1) v = fmaxf(v, __shfl_xor(v, m, 32));
  return v;
}
__device__ __forceinline__ float wave_sum_f(float v) {
#pragma unroll
  for (int m = 16; m > 0; m >>= 1) v += __shfl_xor(v, m, 32);
  return v;
}

// One wave32 per row: lane handles atoms {lane, lane+32}. Softmax via wave
// shuffles; Bellman projection scattered into a per-wave LDS bin buffer with
// ds_add_f32 atomics (bin conflicts within a wave are expected since
// gamma*dz < dz); cross-entropy dot reduced per wave; per-block partial to ws.
__global__ __launch_bounds__(BLOCK_THREADS) void c51_main(
    const float* __restrict__ next_logits,
    const float* __restrict__ old_logits,
    const float* __restrict__ reward,
    const float* __restrict__ done,
    const int*   __restrict__ actions,
    float* __restrict__ out,          // out[0]=loss (kernel2), out[1..]=proj
    float* __restrict__ ws_partials,
    int n_rows)
{
  __shared__ float sh_proj[WAVES_PER_BLOCK][N_ATOMS + 1];
  __shared__ float sh_wacc[WAVES_PER_BLOCK];

  const int lane   = threadIdx.x & 31;
  const int wave   = threadIdx.x >> 5;
  const int gwave  = blockIdx.x * WAVES_PER_BLOCK + wave;
  const int nwaves = gridDim.x * WAVES_PER_BLOCK;

  const float delta_z = (VMAX_C - VMIN_C) / (float)(N_ATOMS - 1);
  const float inv_dz  = 1.0f / delta_z;
  const bool  has2    = (lane < N_ATOMS - 32);   // lanes 0..18 own atom lane+32

  float* shp = sh_proj[wave];
  float  acc = 0.0f;

  for (int row = gwave; row < n_rows; row += nwaves) {
    if (row + nwaves < n_rows) {  // gfx1250 global_prefetch_b8 for next row
      __builtin_prefetch(next_logits + (size_t)(row + nwaves) * N_ATOMS, 0, 3);
      __builtin_prefetch(old_logits  + (size_t)(row + nwaves) * (ACTION_DIM * N_ATOMS), 0, 3);
    }

    // ---- softmax(next_logits[row]) ----
    const float* nl = next_logits + (size_t)row * N_ATOMS;
    float x0 = nl[lane];
    float x1 = has2 ? nl[32 + lane] : -__builtin_inff();
    float mx = wave_max_f(fmaxf(x0, x1));
    float e0 = __expf(x0 - mx);
    float e1 = has2 ? __expf(x1 - mx) : 0.0f;
    float inv_s = 1.0f / wave_sum_f(e0 + e1);
    float p0 = e0 * inv_s;
    float p1 = e1 * inv_s;

    float r    = reward[row];
    float coef = (done[row] > 0.95f) ? 0.0f : GAMMA_C;

    // ---- zero per-wave bin buffer, then scatter with LDS float atomics ----
    shp[lane] = 0.0f;
    if (lane < (N_ATOMS + 1 - 32)) shp[32 + lane] = 0.0f;
    asm volatile("s_wait_dscnt 0" ::: "memory");

    {
      float sup = VMIN_C + delta_z * (float)lane;
      float tz  = fminf(fmaxf(r + coef * sup, VMIN_C), VMAX_C);
      float bb  = (tz - VMIN_C) * inv_dz;
      float lf = floorf(bb), uf = ceilf(bb);          // lf==uf -> both terms 0 (matches ref)
      atomicAdd(&shp[(int)lf], p0 * (uf - bb));
      atomicAdd(&shp[(int)uf], p0 * (bb - lf));
    }
    if (has2) {
      float sup = VMIN_C + delta_z * (float)(32 + lane);
      float tz  = fminf(fmaxf(r + coef * sup, VMIN_C), VMAX_C);
      float bb  = (tz - VMIN_C) * inv_dz;
      float lf = floorf(bb), uf = ceilf(bb);
      atomicAdd(&shp[(int)lf], p1 * (uf - bb));
      atomicAdd(&shp[(int)uf], p1 * (bb - lf));
    }
    asm volatile("s_wait_dscnt 0" ::: "memory");

    float pr0 = shp[lane];
    float pr1 = has2 ? shp[32 + lane] : 0.0f;

    float* po = out + 1 + (size_t)row * N_ATOMS;
    po[lane] = pr0;
    if (has2) po[32 + lane] = pr1;

    // ---- cross-entropy with log_softmax(old_logits[row, action]) ----
    int act = actions[row];
    const float* ol = old_logits + ((size_t)row * ACTION_DIM + (size_t)act) * N_ATOMS;
    float y0 = ol[lane];
    float y1 = has2 ? ol[32 + lane] : -__builtin_inff();
    float m2 = wave_max_f(fmaxf(y0, y1));
    float f0 = __expf(y0 - m2);
    float f1 = has2 ? __expf(y1 - m2) : 0.0f;
    float lse = __logf(wave_sum_f(f0 + f1)) + m2;
    float dot = pr0 * (y0 - lse) + (has2 ? pr1 * (y1 - lse) : 0.0f);
    acc += wave_sum_f(dot);
  }

  if (lane == 0) sh_wacc[wave] = acc;
  __syncthreads();
  if (threadIdx.x == 0) {
    float bsum = 0.0f;
#pragma unroll
    for (int w = 0; w < WAVES_PER_BLOCK; ++w) bsum += sh_wacc[w];
    ws_partials[blockIdx.x] = bsum;
  }
}

// Final reduction with V_WMMA_F32_16X16X4_F32: B = all-ones, so
// D[m][n] = C[m][n] + sum_k A[m][k]. Summing every D element over all lanes
// gives 16 * (sum of all loaded values) regardless of the exact A/D lane
// striping -> layout-robust in a compile-only environment.
__global__ __launch_bounds__(32) void c51_reduce(
    const float* __restrict__ partials, float* __restrict__ out,
    int nparts, float inv_b)
{
  const int lane  = threadIdx.x;
  const int m     = lane & 15;            // A: lane holds row M=lane%16,
  const int kbase = (lane >> 4) << 1;     //    K = 2*(lane/16) + vgpr
  v2f ones; ones.x = 1.0f; ones.y = 1.0f;
  v8f c = {};
  for (int base = 0; base < nparts; base += 64) {
    v2f a;
    a.x = partials[base + m * 4 + kbase + 0];
    a.y = partials[base + m * 4 + kbase + 1];
    c = __builtin_amdgcn_wmma_f32_16x16x4_f32(false, a, false, ones,
                                              (short)0, c, false, false);
  }
  float v = c[0] + c[1] + c[2] + c[3] + c[4] + c[5] + c[6] + c[7];
  float tot16 = wave_sum_f(v);            // = 16 * sum(partials)
  if (lane == 0) out[0] = -(tot16 * (1.0f / 16.0f)) * inv_b;  // loss = -mean
}

extern "C" void kernel_launch(void* const* d_in, const int* in_sizes, int n_in,
                              void* d_out, int out_size, void* d_ws, size_t ws_size,
                              hipStream_t stream) {
  const float* next_logits = (const float*)d_in[0];
  const float* old_logits  = (const float*)d_in[1];
  const float* reward      = (const float*)d_in[2];
  const float* done        = (const float*)d_in[3];
  const int*   actions     = (const int*)d_in[4];
  float* out      = (float*)d_out;
  float* partials = (float*)d_ws;         // NBLK floats = 8 KB of scratch
  const int n_rows = in_sizes[2];         // reward is [B]

  c51_main<<<NBLK, BLOCK_THREADS, 0, stream>>>(
      next_logits, old_logits, reward, done, actions, out, partials, n_rows);
  c51_reduce<<<1, 32, 0, stream>>>(partials, out, NBLK, 1.0f / (float)n_rows);
}